// AttentionLayer_77558519431766
// MI455X (gfx1250) — compile-verified
//
#include <hip/hip_runtime.h>
#include <hip/hip_bf16.h>

// AFM attention layer, algebraically simplified:
//   softmax over a size-1 axis == 1  =>  out[b,e] = sum_{i<j} x[b,i,e]*x[b,j,e]
//                                              = 0.5*( S[b,e]^2 - Q[b,e] )
//   with S = sum_f x[b,f,e], Q = sum_f x[b,f,e]^2.
// The F-reduction is done with V_WMMA_F32_16X16X4_F32:
//   Gram = X^T X accumulated over K-chunks of 4  -> diagonal holds Q
//   Sums = ones(16x4) x X accumulated            -> every row holds S
// Kernel is HBM-bound: ~26.7 MB of traffic -> ~1.15 us at 23.3 TB/s.

typedef __attribute__((ext_vector_type(2))) float v2f;
typedef __attribute__((ext_vector_type(8))) float v8f;

#define AFM_B 2048
#define AFM_F 50
#define AFM_E 64

__global__ __launch_bounds__(256) void afm_pairsum_wmma_kernel(
    const float* __restrict__ x,   // [B, F, E]
    float* __restrict__ out)       // [B, E]
{
    const int lane = threadIdx.x & 31;
    const int waveInBlock = threadIdx.x >> 5;
    const int wave = blockIdx.x * 8 + waveInBlock;   // 8 waves per 256-thread block
    const int b    = wave >> 2;                      // 4 E-tiles of 16 per batch
    const int tile = wave & 3;
    if (b >= AFM_B) return;                          // grid sized exactly; never diverges

    const int e0    = tile * 16;
    const int ecol  = lane & 15;          // N (and M for the A tile) within the 16-wide tile
    const int khalf = (lane >> 4) * 2;    // 32-bit A 16x4 layout: lanes16-31 hold K=2,3

    // Per-lane base pointer into x[b, :, e0+ecol]; stride between features = E floats.
    const float* xb = x + (size_t)b * (AFM_F * AFM_E) + e0 + ecol;

    v8f gram = {};   // diagonal accumulates Q[e]
    v8f sums = {};   // every row accumulates S[e]
    v2f ones; ones.x = 1.0f; ones.y = 1.0f;

    // 13 chunks of K=4 cover F=50 (last chunk zero-padded).
    #pragma unroll
    for (int fc = 0; fc < AFM_F; fc += 4) {
        const int f0 = fc + khalf;
        const int f1 = f0 + 1;
        v2f a;
        a.x = (f0 < AFM_F) ? xb[(size_t)f0 * AFM_E] : 0.0f;
        a.y = (f1 < AFM_F) ? xb[(size_t)f1 * AFM_E] : 0.0f;

        // Gram += A * A  (A registers serve as both A- and B-tiles: X^T X)
        gram = __builtin_amdgcn_wmma_f32_16x16x4_f32(
            /*neg_a=*/false, a, /*neg_b=*/false, a,
            /*c_mod=*/(short)0, gram, /*reuse_a=*/false, /*reuse_b=*/false);
        // Sums += ones * A  -> column sums S[e] replicated in every row
        sums = __builtin_amdgcn_wmma_f32_16x16x4_f32(
            /*neg_a=*/false, ones, /*neg_b=*/false, a,
            /*c_mod=*/(short)0, sums, /*reuse_a=*/false, /*reuse_b=*/false);
    }

    // S[e]: VGPR0 of sums -> lanes 0-15: (M=0, N=lane); lanes 16-31: (M=8, N=lane-16).
    const float S = sums[0];            // lane L holds S[e = L % 16]

    // Q[e] = Gram[e,e]: lanes 0-7 -> vgpr=lane (e=lane); lanes 24-31 -> vgpr=lane-24 (e=lane-16).
    const int qi = lane & 7;
    float Q = gram[0];
    Q = (qi == 1) ? gram[1] : Q;
    Q = (qi == 2) ? gram[2] : Q;
    Q = (qi == 3) ? gram[3] : Q;
    Q = (qi == 4) ? gram[4] : Q;
    Q = (qi == 5) ? gram[5] : Q;
    Q = (qi == 6) ? gram[6] : Q;
    Q = (qi == 7) ? gram[7] : Q;

    const bool lo = (lane < 8);
    const bool hi = (lane >= 24);
    if (lo || hi) {
        const int e = lo ? lane : (lane - 16);
        out[(size_t)b * AFM_E + e0 + e] = 0.5f * (S * S - Q);
    }
}

extern "C" void kernel_launch(void* const* d_in, const int* in_sizes, int n_in,
                              void* d_out, int out_size, void* d_ws, size_t ws_size,
                              hipStream_t stream) {
    const float* x = (const float*)d_in[0];   // inputs [2048, 50, 64]
    // d_in[1..3] (attention weights / projection / bias) are mathematically dead:
    // softmax over the size-1 axis is identically 1, so they never touch the output.
    float* out = (float*)d_out;               // [2048, 64]

    // 2048 batches * 4 E-tiles = 8192 waves; 8 wave32s per 256-thread block.
    const int blocks = (AFM_B * 4) / 8;       // = 1024, exact
    afm_pairsum_wmma_kernel<<<blocks, 256, 0, stream>>>(x, out);
}